// GraphDeformLoss_Neural_Partial_68719477143
// MI455X (gfx1250) — compile-verified
//
#include <hip/hip_runtime.h>
#include <hip/hip_bf16.h>
#include <math.h>

typedef __attribute__((ext_vector_type(2))) float v2f;
typedef __attribute__((ext_vector_type(8))) float v8f;
typedef __attribute__((ext_vector_type(4))) int   v4i;

#define NPTS 4995
#define FC   128
#define KNN  10
#define IMG  224
#define IMGPIX (IMG*IMG)
#define LDSTRIDE 132   // padded row stride (floats) for feat2 tile to dodge bank conflicts

#define AS1 __attribute__((address_space(1)))
#define AS3 __attribute__((address_space(3)))

#if __has_builtin(__builtin_amdgcn_global_load_async_to_lds_b128)
#define HAVE_ASYNC_LDS 1
#else
#define HAVE_ASYNC_LDS 0
#endif

__device__ __forceinline__ void wait_asynccnt0() {
#if __has_builtin(__builtin_amdgcn_s_wait_asynccnt)
  __builtin_amdgcn_s_wait_asynccnt(0);
#else
  asm volatile("s_wait_asynccnt 0x0" ::: "memory");
#endif
}

// ---------- helpers ----------
__device__ __forceinline__ float redsum16(float v) {
  v += __shfl_xor(v, 1, 32);
  v += __shfl_xor(v, 2, 32);
  v += __shfl_xor(v, 4, 32);
  v += __shfl_xor(v, 8, 32);
  return v;
}
__device__ __forceinline__ float redmax16(float v) {
  v = fmaxf(v, __shfl_xor(v, 1, 32));
  v = fmaxf(v, __shfl_xor(v, 2, 32));
  v = fmaxf(v, __shfl_xor(v, 4, 32));
  v = fmaxf(v, __shfl_xor(v, 8, 32));
  return v;
}
__device__ __forceinline__ float blockReduceSum256(float v) {
  __shared__ float sdata[8];
  v += __shfl_xor(v, 1, 32); v += __shfl_xor(v, 2, 32);
  v += __shfl_xor(v, 4, 32); v += __shfl_xor(v, 8, 32);
  v += __shfl_xor(v, 16, 32);
  int w = threadIdx.x >> 5, ln = threadIdx.x & 31;
  if (ln == 0) sdata[w] = v;
  __syncthreads();
  float r = 0.f;
  if (threadIdx.x < 8) r = sdata[threadIdx.x];
  if (threadIdx.x < 32) {
    r += __shfl_xor(r, 1, 32); r += __shfl_xor(r, 2, 32); r += __shfl_xor(r, 4, 32);
  }
  __syncthreads();
  return r;  // valid on thread 0
}
__device__ __forceinline__ unsigned fkey(float f) {
  unsigned u = __float_as_uint(f);
  return (u & 0x80000000u) ? ~u : (u | 0x80000000u);
}
__device__ __forceinline__ float funkey(unsigned k) {
  unsigned u = (k & 0x80000000u) ? (k & 0x7FFFFFFFu) : ~k;
  return __uint_as_float(u);
}
__device__ __forceinline__ float sigmoidf_(float x) { return 1.0f / (1.0f + __expf(-x)); }

// ---------- init: zero images/accumulators, set min/max sentinels ----------
__global__ void init_kernel(float* img, float* accs, unsigned* mm, int nimg, int nmm) {
  int gid = blockIdx.x * blockDim.x + threadIdx.x;
  if (gid < nimg) img[gid] = 0.f;
  if (gid < 3) accs[gid] = 0.f;
  if (gid < nmm) mm[gid] = ((gid & 3) < 2) ? 0xFFFFFFFFu : 0u;  // slots 0,1=min 2,3=max
}

// ---------- squared feature norms ----------
__global__ __launch_bounds__(256) void norms_kernel(const float* __restrict__ feat1,
                                                    const float* __restrict__ feat2,
                                                    float* __restrict__ n1sq,
                                                    float* __restrict__ n2sq, int B) {
  int gid = blockIdx.x * blockDim.x + threadIdx.x;
  int total = B * NPTS;
  int set = gid / total;           // 0 -> feat1, 1 -> feat2
  int i = gid - set * total;
  if (set > 1) return;
  const float4* f = (const float4*)((set == 0 ? feat1 : feat2) + (size_t)i * FC);
  float s = 0.f;
#pragma unroll 8
  for (int j = 0; j < FC / 4; ++j) {
    float4 x = f[j];
    s += x.x * x.x + x.y * x.y + x.z * x.z + x.w * x.w;
  }
  (set == 0 ? n1sq : n2sq)[i] = s;
}

// ---------- pass 1: fused cdist -> softmax -> Pi @ verts2 via FP32 WMMA ----------
// Double-buffered feat2 tile staging through GLOBAL_LOAD_ASYNC_TO_LDS_B128 (ASYNCcnt).
__global__ __launch_bounds__(256)
void pass1_softmax_wmma(const float* __restrict__ feat1, const float* __restrict__ feat2,
                        const float* __restrict__ verts2,
                        const float* __restrict__ n1sq, const float* __restrict__ n2sq,
                        float* __restrict__ verts12) {
  const int b    = blockIdx.y;
  const int wave = threadIdx.x >> 5;
  const int lane = threadIdx.x & 31;
  const int half = lane >> 4;   // K-half for A/B operands
  const int l16  = lane & 15;
  const int n0   = blockIdx.x * 128 + wave * 16;

  __shared__ float lf2[2][16 * LDSTRIDE];

  // Preload A (16 rows x 128 K), fp32 16x16x4 layout: lanes 0-15 hold K=2h..2h+1
  v2f a[32];
  {
    int nrow = n0 + l16; if (nrow >= NPTS) nrow = NPTS - 1;
    const float* fp = feat1 + ((size_t)b * NPTS + nrow) * FC;
#pragma unroll
    for (int j = 0; j < 32; ++j) {
      int k = 4 * j + 2 * half;
      a[j].x = fp[k]; a[j].y = fp[k + 1];
    }
  }
  float n1r[8];
#pragma unroll
  for (int r = 0; r < 8; ++r) {
    int nrow = n0 + r + 8 * half; if (nrow >= NPTS) nrow = NPTS - 1;
    n1r[r] = n1sq[(size_t)b * NPTS + nrow];
  }

  float rmax[8], rsum[8], av[8][3];
#pragma unroll
  for (int r = 0; r < 8; ++r) {
    rmax[r] = -1e30f; rsum[r] = 0.f; av[r][0] = av[r][1] = av[r][2] = 0.f;
  }

#if HAVE_ASYNC_LDS
  // async stage one 16xFC tile into lf2[buf]: 512 x 16B chunks, 2 per thread
  auto issue_tile = [&](int m0, int buf) {
#pragma unroll
    for (int c0 = 0; c0 < 2; ++c0) {
      int c = (int)threadIdx.x + c0 * 256;
      int row = c >> 5, k4 = (c & 31) * 4;
      int mg = m0 + row;
      if (mg < NPTS) {
        const float* gp = feat2 + ((size_t)b * NPTS + mg) * FC + k4;
        float* lp = &lf2[buf][row * LDSTRIDE + k4];
        __builtin_amdgcn_global_load_async_to_lds_b128(
            (AS1 v4i*)gp, (AS3 v4i*)lp, 0, 0);
      }
    }
  };
  issue_tile(0, 0);
#endif

  int it = 0;
  for (int m0 = 0; m0 < NPTS; m0 += 16, ++it) {
    const int cur = it & 1;
#if HAVE_ASYNC_LDS
    wait_asynccnt0();       // our buffer's async copies have landed in LDS
    __syncthreads();        // everyone done with the buffer we are about to overwrite
    if (m0 + 16 < NPTS) issue_tile(m0 + 16, cur ^ 1);
#else
    __syncthreads();
    for (int t = threadIdx.x; t < 16 * FC; t += 256) {
      int mr = t >> 7, k = t & 127;
      int mg = m0 + mr;
      lf2[cur][mr * LDSTRIDE + k] = (mg < NPTS) ? feat2[((size_t)b * NPTS + mg) * FC + k] : 0.f;
    }
    __syncthreads();
#endif

    v8f acc = {};
#pragma unroll
    for (int j = 0; j < 32; ++j) {
      int k = 4 * j + 2 * half;
      v2f bv; bv.x = lf2[cur][l16 * LDSTRIDE + k]; bv.y = lf2[cur][l16 * LDSTRIDE + k + 1];
      acc = __builtin_amdgcn_wmma_f32_16x16x4_f32(false, a[j], false, bv,
                                                  (short)0, acc, false, false);
    }

    // per-lane column data (clamped loads; masked below)
    const int mcol = m0 + l16;
    const bool colok = mcol < NPTS;
    const int mc = colok ? mcol : NPTS - 1;
    const float vx = verts2[((size_t)b * NPTS + mc) * 3 + 0];
    const float vy = verts2[((size_t)b * NPTS + mc) * 3 + 1];
    const float vz = verts2[((size_t)b * NPTS + mc) * 3 + 2];
    const float n2c = n2sq[(size_t)b * NPTS + mc];

#pragma unroll
    for (int r = 0; r < 8; ++r) {
      float sq = fmaxf(n1r[r] + n2c - 2.0f * acc[r], 1e-12f);
      float s = colok ? (-100.0f * sqrtf(sq)) : -1e30f;
      float tmax = redmax16(s);
      float nm = fmaxf(rmax[r], tmax);
      float scale = __expf(rmax[r] - nm);
      float p = colok ? __expf(s - nm) : 0.0f;
      float ps = redsum16(p);
      float px = redsum16(p * vx);
      float py = redsum16(p * vy);
      float pz = redsum16(p * vz);
      rsum[r]  = rsum[r]  * scale + ps;
      av[r][0] = av[r][0] * scale + px;
      av[r][1] = av[r][1] * scale + py;
      av[r][2] = av[r][2] * scale + pz;
      rmax[r] = nm;
    }
  }

  if (l16 == 0) {
#pragma unroll
    for (int r = 0; r < 8; ++r) {
      int n = n0 + r + 8 * half;
      if (n < NPTS) {
        float inv = 1.0f / rsum[r];
        verts12[((size_t)b * NPTS + n) * 3 + 0] = av[r][0] * inv;
        verts12[((size_t)b * NPTS + n) * 3 + 1] = av[r][1] * inv;
        verts12[((size_t)b * NPTS + n) * 3 + 2] = av[r][2] * inv;
      }
    }
  }
}

// ---------- self reconstruction loss: mean_n min_m ||verts12[n]-verts2[m]||^2 ----------
__global__ __launch_bounds__(256)
void rec_loss_kernel(const float* __restrict__ verts12, const float* __restrict__ verts2,
                     float* __restrict__ accs) {
  const int b = blockIdx.y;
  const int n = blockIdx.x * 256 + threadIdx.x;
  __shared__ float tile[256 * 3];
  float px = 0, py = 0, pz = 0;
  const bool valid = n < NPTS;
  if (valid) {
    px = verts12[((size_t)b * NPTS + n) * 3 + 0];
    py = verts12[((size_t)b * NPTS + n) * 3 + 1];
    pz = verts12[((size_t)b * NPTS + n) * 3 + 2];
  }
  float best = 1e30f;
  for (int m0 = 0; m0 < NPTS; m0 += 256) {
    __syncthreads();
    int mg = m0 + threadIdx.x;
    if (mg < NPTS) {
      tile[threadIdx.x * 3 + 0] = verts2[((size_t)b * NPTS + mg) * 3 + 0];
      tile[threadIdx.x * 3 + 1] = verts2[((size_t)b * NPTS + mg) * 3 + 1];
      tile[threadIdx.x * 3 + 2] = verts2[((size_t)b * NPTS + mg) * 3 + 2];
    }
    __syncthreads();
    int cnt = min(256, NPTS - m0);
    for (int t = 0; t < cnt; ++t) {
      float dx = px - tile[t * 3 + 0];
      float dy = py - tile[t * 3 + 1];
      float dz = pz - tile[t * 3 + 2];
      best = fminf(best, dx * dx + dy * dy + dz * dz);
    }
  }
  float v = valid ? best : 0.f;
  float s = blockReduceSum256(v);
  if (threadIdx.x == 0) atomicAdd(&accs[0], s);
}

// ---------- kNN on verts1 (k=10) + feature deformation loss ----------
__global__ __launch_bounds__(256)
void knn_feat_kernel(const float* __restrict__ verts1, const float* __restrict__ feat1,
                     float* __restrict__ accs) {
  const int b = blockIdx.y;
  const int n = blockIdx.x * 256 + threadIdx.x;
  __shared__ float tile[256 * 3];
  float px = 0, py = 0, pz = 0;
  const bool valid = n < NPTS;
  if (valid) {
    px = verts1[((size_t)b * NPTS + n) * 3 + 0];
    py = verts1[((size_t)b * NPTS + n) * 3 + 1];
    pz = verts1[((size_t)b * NPTS + n) * 3 + 2];
  }
  float bd[KNN]; int bidx[KNN];
#pragma unroll
  for (int t = 0; t < KNN; ++t) { bd[t] = 1e30f; bidx[t] = 0; }

  for (int m0 = 0; m0 < NPTS; m0 += 256) {
    __syncthreads();
    int mg = m0 + threadIdx.x;
    if (mg < NPTS) {
      tile[threadIdx.x * 3 + 0] = verts1[((size_t)b * NPTS + mg) * 3 + 0];
      tile[threadIdx.x * 3 + 1] = verts1[((size_t)b * NPTS + mg) * 3 + 1];
      tile[threadIdx.x * 3 + 2] = verts1[((size_t)b * NPTS + mg) * 3 + 2];
    }
    __syncthreads();
    int cnt = min(256, NPTS - m0);
    for (int t = 0; t < cnt; ++t) {
      float dx = px - tile[t * 3 + 0];
      float dy = py - tile[t * 3 + 1];
      float dz = pz - tile[t * 3 + 2];
      float d = dx * dx + dy * dy + dz * dz;
      if (d < bd[KNN - 1]) {
        bd[KNN - 1] = d; bidx[KNN - 1] = m0 + t;
#pragma unroll
        for (int q = KNN - 1; q > 0; --q) {
          if (bd[q] < bd[q - 1]) {
            float td = bd[q]; bd[q] = bd[q - 1]; bd[q - 1] = td;
            int ti = bidx[q]; bidx[q] = bidx[q - 1]; bidx[q - 1] = ti;
          }
        }
      }
    }
  }

  float fsum = 0.f;
  if (valid) {
    const float4* myf = (const float4*)(feat1 + ((size_t)b * NPTS + n) * FC);
#pragma unroll
    for (int t = 0; t < KNN; ++t) {
      const float4* nf = (const float4*)(feat1 + ((size_t)b * NPTS + bidx[t]) * FC);
      for (int j = 0; j < FC / 4; ++j) {
        float4 x = myf[j], y = nf[j];
        float dx = y.x - x.x, dy = y.y - x.y, dz = y.z - x.z, dw = y.w - x.w;
        fsum += dx * dx + dy * dy + dz * dz + dw * dw;
      }
    }
  }
  float s = blockReduceSum256(fsum);
  if (threadIdx.x == 0) atomicAdd(&accs[1], s);
}

// ---------- per-(batch,set) min/max of x,y via ordered-uint atomics ----------
__global__ __launch_bounds__(256)
void minmax_kernel(const float* __restrict__ verts12, const float* __restrict__ verts2,
                   unsigned* __restrict__ mm) {
  const int bs = blockIdx.y;                // b*2 + set
  const int b = bs >> 1, set = bs & 1;
  const float* pts = (set == 0) ? (verts12 + (size_t)b * NPTS * 3)
                                : (verts2 + (size_t)b * NPTS * 3);
  const int n = blockIdx.x * 256 + threadIdx.x;
  if (n >= NPTS) return;
  float x = pts[n * 3 + 0], y = pts[n * 3 + 1];
  atomicMin(&mm[bs * 4 + 0], fkey(x));
  atomicMin(&mm[bs * 4 + 1], fkey(y));
  atomicMax(&mm[bs * 4 + 2], fkey(x));
  atomicMax(&mm[bs * 4 + 3], fkey(y));
}

// ---------- per-(batch,set) projection scalars ----------
__global__ void proj_scalars_kernel(const unsigned* __restrict__ mm, float* __restrict__ scal,
                                    int nbs) {
  int bs = blockIdx.x * blockDim.x + threadIdx.x;
  if (bs >= nbs) return;
  float minx = funkey(mm[bs * 4 + 0]), miny = funkey(mm[bs * 4 + 1]);
  float maxx = funkey(mm[bs * 4 + 2]), maxy = funkey(mm[bs * 4 + 3]);
  float gs = fmaxf(fmaxf(maxx - minx, maxy - miny) / (float)(IMG - 3), 1e-12f);
  float maxix = floorf((maxx - minx) / gs);
  float maxiy = floorf((maxy - miny) / gs);
  // idx_dense range: [minidx-2+1, maxidx+2+1] with minidx=0
  float cx = floorf((maxix + 2.0f) * 0.5f);
  float cy = floorf((maxiy + 2.0f) * 0.5f);
  scal[bs * 8 + 0] = gs;
  scal[bs * 8 + 1] = minx;
  scal[bs * 8 + 2] = miny;
  scal[bs * 8 + 3] = (float)IMG * 0.5f - cx - 1.0f;   // offset x
  scal[bs * 8 + 4] = (float)IMG * 0.5f - cy - 1.0f;   // offset y
}

// ---------- scatter-add projection (single channel; channels are identical) ----------
__global__ __launch_bounds__(256)
void scatter_kernel(const float* __restrict__ verts12, const float* __restrict__ verts2,
                    const float* __restrict__ scal, float* __restrict__ img) {
  const int bs = blockIdx.y;
  const int b = bs >> 1, set = bs & 1;
  const float* pts = (set == 0) ? (verts12 + (size_t)b * NPTS * 3)
                                : (verts2 + (size_t)b * NPTS * 3);
  const int n = blockIdx.x * 256 + threadIdx.x;
  if (n >= NPTS) return;
  float x = pts[n * 3 + 0], y = pts[n * 3 + 1], z = pts[n * 3 + 2];
  float gs = scal[bs * 8 + 0], minx = scal[bs * 8 + 1], miny = scal[bs * 8 + 2];
  float offx = scal[bs * 8 + 3], offy = scal[bs * 8 + 4];
  float ix = floorf((x - minx) / gs);
  float iy = floorf((y - miny) / gs);
  float* im = img + (size_t)bs * IMGPIX;
#pragma unroll
  for (int oi = -2; oi <= 2; ++oi) {
#pragma unroll
    for (int oj = -2; oj <= 2; ++oj) {
      float fx = ix + (float)oi + 1.0f + offx;
      float fy = iy + (float)oj + 1.0f + offy;
      fx += (fx < 0.f ? 1.f : 0.f) - (fx > (float)(IMG - 1) ? 1.f : 0.f);
      fy += (fy < 0.f ? 1.f : 0.f) - (fy > (float)(IMG - 1) ? 1.f : 0.f);
      fx = fminf(fmaxf(fx, 0.f), (float)(IMG - 1));
      fy = fminf(fmaxf(fy, 0.f), (float)(IMG - 1));
      int fl = (int)(fx * (float)IMG + fy);
      atomicAdd(&im[fl], z);
    }
  }
}

// ---------- image MSE (sigmoid space) ----------
__global__ __launch_bounds__(256)
void img_loss_kernel(const float* __restrict__ img, float* __restrict__ accs, int total) {
  int i = blockIdx.x * 256 + threadIdx.x;
  float v = 0.f;
  if (i < total) {
    int b = i / IMGPIX, pix = i - b * IMGPIX;
    float a1 = img[(size_t)(b * 2 + 0) * IMGPIX + pix];
    float a2 = img[(size_t)(b * 2 + 1) * IMGPIX + pix];
    float d = sigmoidf_(a1) - sigmoidf_(a2);
    v = d * d;
  }
  float s = blockReduceSum256(v);
  if (threadIdx.x == 0) atomicAdd(&accs[2], s);
}

// ---------- finalize ----------
__global__ void finalize_kernel(const float* __restrict__ accs, float* __restrict__ out, int B) {
  if (threadIdx.x == 0 && blockIdx.x == 0) {
    float rec  = accs[0] / ((float)B * (float)NPTS);
    float feat = accs[1] / ((float)B * (float)NPTS * (float)KNN * (float)FC);
    float img  = accs[2] / ((float)B * (float)IMGPIX);  // 3 identical channels cancel
    out[0] = rec + feat + img;
  }
}

extern "C" void kernel_launch(void* const* d_in, const int* in_sizes, int n_in,
                              void* d_out, int out_size, void* d_ws, size_t ws_size,
                              hipStream_t stream) {
  const float* verts1 = (const float*)d_in[0];
  const float* verts2 = (const float*)d_in[1];
  const float* feat1  = (const float*)d_in[2];
  const float* feat2  = (const float*)d_in[3];
  const int B = in_sizes[0] / (NPTS * 3);

  // workspace layout (float-aligned)
  float* ws = (float*)d_ws;
  size_t off = 0;
  float* verts12 = ws + off; off += (size_t)B * NPTS * 3;
  float* n1sq    = ws + off; off += (size_t)B * NPTS;
  float* n2sq    = ws + off; off += (size_t)B * NPTS;
  float* scal    = ws + off; off += (size_t)B * 2 * 8;
  float* accs    = ws + off; off += 64;
  unsigned* mm   = (unsigned*)(ws + off); off += 64;
  float* img     = ws + off; off += (size_t)B * 2 * IMGPIX;
  (void)ws_size; (void)n_in; (void)out_size;

  const int nimg = B * 2 * IMGPIX;
  const int nmm = B * 2 * 4;
  init_kernel<<<(nimg + 255) / 256, 256, 0, stream>>>(img, accs, mm, nimg, nmm);

  norms_kernel<<<(B * NPTS * 2 + 255) / 256, 256, 0, stream>>>(feat1, feat2, n1sq, n2sq, B);

  dim3 g1((NPTS + 127) / 128, B);
  pass1_softmax_wmma<<<g1, 256, 0, stream>>>(feat1, feat2, verts2, n1sq, n2sq, verts12);

  dim3 g2((NPTS + 255) / 256, B);
  rec_loss_kernel<<<g2, 256, 0, stream>>>(verts12, verts2, accs);
  knn_feat_kernel<<<g2, 256, 0, stream>>>(verts1, feat1, accs);

  dim3 g3((NPTS + 255) / 256, B * 2);
  minmax_kernel<<<g3, 256, 0, stream>>>(verts12, verts2, mm);
  proj_scalars_kernel<<<1, 32, 0, stream>>>(mm, scal, B * 2);
  scatter_kernel<<<g3, 256, 0, stream>>>(verts12, verts2, scal, img);

  const int tot = B * IMGPIX;
  img_loss_kernel<<<(tot + 255) / 256, 256, 0, stream>>>(img, accs, tot);

  finalize_kernel<<<1, 32, 0, stream>>>(accs, (float*)d_out, B);
}